// GE2ELoss_65446711656459
// MI455X (gfx1250) — compile-verified
//
#include <hip/hip_runtime.h>

typedef __attribute__((ext_vector_type(2))) float v2f;
typedef __attribute__((ext_vector_type(8))) float v8f;

#define N_SPK 2048
#define M_UTT 16
#define D_DIM 128
#define ROWS (N_SPK * M_UTT)

// ---------------------------------------------------------------------------
// Kernel 1: centroids[n][d] = mean_m x[n][m][d]; also zero the scalar output
// (graph-replay safe: runs first in-stream on every call).
// ---------------------------------------------------------------------------
__global__ void ge2e_centroid_kernel(const float* __restrict__ x,
                                     float* __restrict__ cent,
                                     float* __restrict__ out) {
    const int n = blockIdx.x;
    const int d = threadIdx.x;
    const float* xp = x + (size_t)n * M_UTT * D_DIM + d;
    float s = 0.0f;
#pragma unroll
    for (int m = 0; m < M_UTT; ++m) s += xp[m * D_DIM];
    cent[n * D_DIM + d] = s * (1.0f / M_UTT);
    if (n == 0 && d == 0) out[0] = 0.0f;
}

// ---------------------------------------------------------------------------
// Kernel 2: diag[r] = x_r . ((M*c_n - x_r)/(M-1))  (leave-one-out similarity)
// ---------------------------------------------------------------------------
__global__ void ge2e_diag_kernel(const float* __restrict__ x,
                                 const float* __restrict__ cent,
                                 float* __restrict__ diag) {
    const int r = blockIdx.x;        // row = n*M + m
    const int n = r >> 4;
    const int d = threadIdx.x;
    const float xv = x[(size_t)r * D_DIM + d];
    const float cv = cent[n * D_DIM + d];
    const float ex = (16.0f * cv - xv) * (1.0f / 15.0f);
    __shared__ float red[D_DIM];
    red[d] = xv * ex;
    __syncthreads();
#pragma unroll
    for (int s = D_DIM / 2; s > 0; s >>= 1) {
        if (d < s) red[d] += red[d + s];
        __syncthreads();
    }
    if (d == 0) diag[r] = red[0];
}

// ---------------------------------------------------------------------------
// Kernel 3: fused GEMM (WMMA f32 16x16x4) + sum-exp + log-softmax loss.
// One block per speaker n (16 rows). 8 waves; wave w owns col tiles w, w+8, ...
// Logits are bounded (|sim|<=1.14, w=10, b=-5) so plain sum-exp is safe.
// ---------------------------------------------------------------------------
__launch_bounds__(256, 1)
__global__ void ge2e_main_kernel(const float* __restrict__ x,
                                 const float* __restrict__ cent,
                                 const float* __restrict__ diag,
                                 const float* __restrict__ wp,
                                 const float* __restrict__ bp,
                                 float* __restrict__ out) {
    __shared__ float xtile[16 * D_DIM];   // 8 KB row block of x
    __shared__ float psum[8][16];         // per-wave partial sum-exp per row
    __shared__ float rowloss[16];

    const int tid  = threadIdx.x;
    const int wave = tid >> 5;
    const int lane = tid & 31;
    const int h    = lane >> 4;           // half-wave (selects K pair / row set)
    const int c    = lane & 15;           // column-in-tile (B/D), row (A)
    const int nOwn = blockIdx.x;          // this block's speaker id
    const int row_base = nOwn * 16;
    const float w = wp[0];
    const float b = bp[0];

    // Stage the 16x128 row block of x into LDS (contiguous, coalesced).
    const float* xblk = x + (size_t)row_base * D_DIM;
#pragma unroll
    for (int i = 0; i < 8; ++i) xtile[tid + i * 256] = xblk[tid + i * 256];
    __syncthreads();

    float sexp[8];
#pragma unroll
    for (int v = 0; v < 8; ++v) sexp[v] = 0.0f;

    const float* arow = xtile + c * D_DIM;   // A operand: row M = lane%16

    // Each wave walks 16 of the 128 column tiles (16 centroids each).
    for (int t = wave; t < N_SPK / 16; t += 8) {
        const int col_base = t * 16;
        const float* brow = cent + (size_t)(col_base + c) * D_DIM; // B: col N = lane%16

        v8f acc = {};
#pragma unroll
        for (int k0 = 0; k0 < D_DIM; k0 += 4) {
            // ISA 7.12.2: 32-bit A 16x4 -> lane holds K = k0+2h, k0+2h+1 of row lane%16.
            // B 4x16 symmetric: lane holds K = k0+2h, k0+2h+1 of column lane%16.
            v2f a, bb;
            a.x  = arow[k0 + 2 * h];
            a.y  = arow[k0 + 2 * h + 1];
            bb.x = brow[k0 + 2 * h];
            bb.y = brow[k0 + 2 * h + 1];
            acc = __builtin_amdgcn_wmma_f32_16x16x4_f32(
                /*neg_a=*/false, a, /*neg_b=*/false, bb,
                /*c_mod=*/(short)0, acc, /*reuse_a=*/false, /*reuse_b=*/false);
        }

        // D layout: lane c = column, VGPR v = row (v + 8*h). Apply scale/bias,
        // override the own-speaker column with the leave-one-out logit.
        const int gc = col_base + c;
#pragma unroll
        for (int v = 0; v < 8; ++v) {
            const int m = v + 8 * h;
            float logit = (gc == nOwn) ? fmaf(w, diag[row_base + m], b)
                                       : fmaf(w, acc[v], b);
            sexp[v] += __expf(logit);
        }
    }

    // Reduce each row's partial sum across the 16 lanes of this half-wave
    // (xor masks 1,2,4,8 stay within a 16-lane group in wave32).
#pragma unroll
    for (int v = 0; v < 8; ++v) {
        float s = sexp[v];
        s += __shfl_xor(s, 1, 32);
        s += __shfl_xor(s, 2, 32);
        s += __shfl_xor(s, 4, 32);
        s += __shfl_xor(s, 8, 32);
        sexp[v] = s;
    }
    if (c == 0) {
#pragma unroll
        for (int v = 0; v < 8; ++v) psum[wave][v + 8 * h] = sexp[v];
    }
    __syncthreads();

    // Combine 8 wave partials per row; loss_r = log(sum exp) - diag logit.
    if (tid < 16) {
        float S = 0.0f;
#pragma unroll
        for (int wv = 0; wv < 8; ++wv) S += psum[wv][tid];
        const float dl = fmaf(w, diag[row_base + tid], b);
        rowloss[tid] = __logf(S) - dl;
    }
    __syncthreads();

    if (tid == 0) {
        float s = 0.0f;
#pragma unroll
        for (int i = 0; i < 16; ++i) s += rowloss[i];
        atomicAdd(out, s * (1.0f / ROWS));
    }
}

// ---------------------------------------------------------------------------
extern "C" void kernel_launch(void* const* d_in, const int* in_sizes, int n_in,
                              void* d_out, int out_size, void* d_ws, size_t ws_size,
                              hipStream_t stream) {
    const float* x = (const float*)d_in[0];
    const float* w = (const float*)d_in[1];
    const float* b = (const float*)d_in[2];
    float* out = (float*)d_out;

    float* cent = (float*)d_ws;                 // 2048*128 f32 = 1 MB
    float* diag = cent + (size_t)N_SPK * D_DIM; // 32768  f32 = 128 KB

    ge2e_centroid_kernel<<<N_SPK, D_DIM, 0, stream>>>(x, cent, out);
    ge2e_diag_kernel<<<ROWS, D_DIM, 0, stream>>>(x, cent, diag);
    ge2e_main_kernel<<<N_SPK, 256, 0, stream>>>(x, cent, diag, w, b, out);
}